// OptimalTransport_23819888623717
// MI455X (gfx1250) — compile-verified
//
#include <hip/hip_runtime.h>

// Sinkhorn optimal transport, one workgroup (256 threads = 8 wave32) per sample.
// S=2048, B=64, N=256, eps=0.01.
//  - C (64x256 f32) DMA'd into LDS by the Tensor Data Mover (tensor_load_to_lds),
//    using TDM LDS-padding to produce a stride-257 (bank-conflict-free) layout.
//  - mu/nu staged with global_load_async_to_lds_b32 (ASYNCcnt).
//  - exp-sum reductions on the matrix pipe via V_WMMA_F32_16X16X4_F32 with an
//    all-ones operand (full f32 precision); maxes + transcendentals on VALU.

#define B_DIM 64
#define N_DIM 256
#define LDS_STRIDE 257   // +1 pad -> bank-conflict-free row AND column walks
#define EPS_F 0.01f
#define INV_EPS 100.0f
#define MAX_ITER 100
#define THRESH 0.1f
#define LOG_EPS 1e-8f

typedef float v2f __attribute__((ext_vector_type(2)));
typedef float v8f __attribute__((ext_vector_type(8)));
typedef unsigned int u32;
typedef u32 v4u __attribute__((ext_vector_type(4)));
typedef u32 v8u __attribute__((ext_vector_type(8)));

__global__ __launch_bounds__(256) void sinkhorn_ot_kernel(
    const float* __restrict__ mu, const float* __restrict__ nu,
    const float* __restrict__ C, float* __restrict__ out)
{
  extern __shared__ float sC[];                 // [B_DIM][LDS_STRIDE] = 65792 B
  __shared__ float su[B_DIM], smu[B_DIM], slogmu[B_DIM];
  __shared__ float sv[N_DIM], slognu[N_DIM];
  __shared__ float srowmax[B_DIM];
  __shared__ float srowpart[4][B_DIM];
  __shared__ float srowsum[2][B_DIM];
  __shared__ float scolmax[N_DIM], scolsum[N_DIM];
  __shared__ float serr[B_DIM];
  __shared__ float serr_tot;

  const int tid  = threadIdx.x;
  const int s    = blockIdx.x;
  const int lane = tid & 31;
  const int wave = tid >> 5;

  // ---- TDM: DMA C (64x256 f32, row major) into LDS, hardware-padded to stride 257 ----
  if (wave == 0) {
    u32 lds_base = (u32)(size_t)(void*)sC;      // low 32 bits of generic ptr = LDS offset
    unsigned long long ga = (unsigned long long)(size_t)C;
    v4u g0;
    g0[0] = 1u;                                  // count=1, user descriptor, no gather
    g0[1] = lds_base;                            // lds_addr
    g0[2] = (u32)ga;                             // global_addr[31:0]
    g0[3] = ((u32)(ga >> 32) & 0x01FFFFFFu)      // global_addr[56:32]
            | (2u << 30);                        // type = 2 ("image")
    v8u g1;
    g1[0] = (2u << 16)                           // data_size = 4 bytes
          | (1u << 20)                           // pad_enable
          | (7u << 22);                          // pad_interval: every 256 DWORDs
                                                 // pad_amount field 0 => 1 DWORD pad
    g1[1] = ((u32)N_DIM) << 16;                  // tensor_dim0[15:0] = 256
    g1[2] = ((u32)B_DIM) << 16;                  // tensor_dim0[31:16]=0 | tensor_dim1[15:0]=64
    g1[3] = ((u32)N_DIM) << 16;                  // tensor_dim1[31:16]=0 | tile_dim0 = 256
    g1[4] = (u32)B_DIM;                          // tile_dim1 = 64, tile_dim2 = 0
    g1[5] = (u32)N_DIM;                          // tensor_dim0_stride[31:0] = 256
    g1[6] = 0u;                                  // stride hi / tensor_dim1_stride lo
    g1[7] = 0u;
    asm volatile("tensor_load_to_lds %0, %1" :: "s"(g0), "s"(g1) : "memory");
  }

  // ---- async global->LDS staging of nu (raw into slognu slot) and mu ----
  {
    u32 dst  = (u32)(size_t)(void*)&slognu[tid];
    u32 voff = (u32)((s * N_DIM + tid) * sizeof(float));
    asm volatile("global_load_async_to_lds_b32 %0, %1, %2"
                 :: "v"(dst), "v"(voff), "s"(nu) : "memory");
  }
  if (tid < B_DIM) {
    u32 dst  = (u32)(size_t)(void*)&smu[tid];
    u32 voff = (u32)((s * B_DIM + tid) * sizeof(float));
    asm volatile("global_load_async_to_lds_b32 %0, %1, %2"
                 :: "v"(dst), "v"(voff), "s"(mu) : "memory");
  }
  asm volatile("s_wait_asynccnt 0x0" ::: "memory");
  if (wave == 0) {
#if __has_builtin(__builtin_amdgcn_s_wait_tensorcnt)
    __builtin_amdgcn_s_wait_tensorcnt(0);
#else
    asm volatile("s_wait_tensorcnt 0x0" ::: "memory");
#endif
  }
  __syncthreads();

  {
    float n = slognu[tid];                       // raw nu staged by async load
    sv[tid]     = 0.0f;
    slognu[tid] = __logf(n + LOG_EPS);
  }
  if (tid < B_DIM) {
    float m = smu[tid];                          // raw mu staged by async load
    su[tid]     = 0.0f;
    slogmu[tid] = __logf(m + LOG_EPS);
  }
  __syncthreads();

  float err = 3.4e38f;
  int it = 0;
  while (it < MAX_ITER && err >= THRESH) {
    // ---------- Phase A: row logsumexp over j, update u (uses old u, old v) ----------
    {
      int r = tid & 63, q = tid >> 6;
      float u_r = su[r];
      float mx = -3.4e38f;
      const int base = r * LDS_STRIDE + q * 64;
      for (int jj = 0; jj < 64; ++jj) {
        float m = (u_r + sv[q * 64 + jj] - sC[base + jj]) * INV_EPS;
        mx = fmaxf(mx, m);
      }
      srowpart[q][r] = mx;
    }
    __syncthreads();
    if (tid < B_DIM) {
      srowmax[tid] = fmaxf(fmaxf(srowpart[0][tid], srowpart[1][tid]),
                           fmaxf(srowpart[2][tid], srowpart[3][tid]));
    }
    __syncthreads();

    // rowsum = E * ones : E tile is the A matrix (16x4 per step), ones is B.
    // Wave w: row tile t=w&3 (16 rows), K half h=w>>2 (128 of 256 columns).
    {
      const int t = wave & 3, h = wave >> 2;
      const int mrow = t * 16 + (lane & 15);     // A layout: lanes 0-15 M=l, 16-31 M=l-16
      const int kofs = (lane >> 4) * 2;          // lanes 0-15 hold K0,K1; 16-31 hold K2,K3
      const float u_m  = su[mrow];
      const float rmax = srowmax[mrow];
      const v2f ones = {1.0f, 1.0f};
      v8f acc = {0.f, 0.f, 0.f, 0.f, 0.f, 0.f, 0.f, 0.f};
      const int rowbase = mrow * LDS_STRIDE;
      #pragma unroll 4
      for (int k0 = h * 128; k0 < h * 128 + 128; k0 += 4) {
        int j0 = k0 + kofs;
        v2f a;
        a.x = __expf((u_m + sv[j0]     - sC[rowbase + j0])     * INV_EPS - rmax);
        a.y = __expf((u_m + sv[j0 + 1] - sC[rowbase + j0 + 1]) * INV_EPS - rmax);
        acc = __builtin_amdgcn_wmma_f32_16x16x4_f32(false, a, false, ones,
                                                    (short)0, acc, false, false);
      }
      // D layout: VGPR r = rows {r, r+8}; every column equal (B = ones).
      if ((lane & 15) == 0) {
        int base = t * 16 + (lane >> 4) * 8;
        #pragma unroll
        for (int r = 0; r < 8; ++r) srowsum[h][base + r] = acc[r];
      }
    }
    __syncthreads();

    if (tid < B_DIM) {
      float rs  = srowsum[0][tid] + srowsum[1][tid];
      float lse = srowmax[tid] + __logf(rs);
      float uo  = su[tid];
      float un  = EPS_F * (slogmu[tid] - lse) + uo;
      su[tid]   = un;
      serr[tid] = fabsf(un - uo);
    }
    __syncthreads();

    // ---------- Phase B: column logsumexp over b, update v (uses NEW u, old v) ----------
    {
      float v_j = sv[tid];
      float mx = -3.4e38f;
      #pragma unroll 8
      for (int b = 0; b < B_DIM; ++b)
        mx = fmaxf(mx, (su[b] + v_j - sC[b * LDS_STRIDE + tid]) * INV_EPS);
      scolmax[tid] = mx;
    }
    if (tid == 0) {                 // per-sample error (reference uses global mean)
      float e = 0.f;
      for (int b = 0; b < B_DIM; ++b) e += serr[b];
      serr_tot = e;
    }
    __syncthreads();

    // colsum = ones^T * E : ones is A, E tile is the B matrix (4x16 per step).
    // Wave w handles column tiles 2w and 2w+1; K = 64 rows in steps of 4.
    {
      const v2f ones = {1.0f, 1.0f};
      const int kofs = (lane >> 4) * 2;          // B layout: lanes 0-15 K0,K1; 16-31 K2,K3
      for (int tt = 0; tt < 2; ++tt) {
        int tile = wave * 2 + tt;
        int n = tile * 16 + (lane & 15);         // lanes 0-15 N=l, 16-31 N=l-16
        float v_n  = sv[n];
        float cmax = scolmax[n];
        v8f acc = {0.f, 0.f, 0.f, 0.f, 0.f, 0.f, 0.f, 0.f};
        #pragma unroll 4
        for (int k0 = 0; k0 < B_DIM; k0 += 4) {
          int b = k0 + kofs;
          v2f bb;
          bb.x = __expf((su[b]     + v_n - sC[b * LDS_STRIDE + n])       * INV_EPS - cmax);
          bb.y = __expf((su[b + 1] + v_n - sC[(b + 1) * LDS_STRIDE + n]) * INV_EPS - cmax);
          acc = __builtin_amdgcn_wmma_f32_16x16x4_f32(false, ones, false, bb,
                                                      (short)0, acc, false, false);
        }
        if (lane < 16) scolsum[n] = acc[0];      // row 0 of D = column sums
      }
    }
    __syncthreads();

    {
      float cs  = scolsum[tid];
      float lse = scolmax[tid] + __logf(cs);
      sv[tid] = EPS_F * (slognu[tid] - lse) + sv[tid];
    }
    __syncthreads();
    err = serr_tot;
    ++it;
  }

  // out[s, j] = sum_b mu[b] * exp((u[b] + v[j] - C[b,j]) / eps)
  {
    float v_j = sv[tid];
    float acc = 0.f;
    #pragma unroll 8
    for (int b = 0; b < B_DIM; ++b)
      acc += smu[b] * __expf((su[b] + v_j - sC[b * LDS_STRIDE + tid]) * INV_EPS);
    out[(size_t)s * N_DIM + tid] = acc;
  }
}

extern "C" void kernel_launch(void* const* d_in, const int* in_sizes, int n_in,
                              void* d_out, int out_size, void* d_ws, size_t ws_size,
                              hipStream_t stream) {
  (void)n_in; (void)out_size; (void)d_ws; (void)ws_size;
  const float* mu = (const float*)d_in[0];   // (S, 64)
  const float* nu = (const float*)d_in[1];   // (S, 256)
  const float* C  = (const float*)d_in[2];   // (64, 256)
  float* out = (float*)d_out;                // (S, 256)
  const int S = in_sizes[0] / B_DIM;
  const size_t lds_bytes = (size_t)B_DIM * LDS_STRIDE * sizeof(float); // 65792
  hipLaunchKernelGGL(sinkhorn_ot_kernel, dim3(S), dim3(256), lds_bytes, stream,
                     mu, nu, C, out);
}